// GetQ_87909390615220
// MI455X (gfx1250) — compile-verified
//
#include <hip/hip_runtime.h>
#include <math.h>

// Problem constants (from reference)
#define N_ATOMS 30
#define NPAIR   435          // 30*29/2
#define NR      91           // 3*N_ATOMS + 1 rows of eij_FE
#define NRP     96           // padded to 6 x 16 WMMA row tiles
#define KC      64           // K-chunk (pairs per chunk)
#define CS      65           // chunk LDS row stride (floats, padded)
#define GS      97           // G LDS row stride (floats, padded)
#define NCHUNK  7            // 7*64 = 448 >= 435 (tail zero-padded)
#define NTHREADS 192         // 6 wave32 waves; wave w owns G row-block w

typedef float v2f __attribute__((ext_vector_type(2)));
typedef float v8f __attribute__((ext_vector_type(8)));

// p -> (i,j), i > j, lower-triangle row-major: p = i*(i-1)/2 + j
__device__ inline void pair_from_index(int p, int& i, int& j) {
    i = (int)((1.0f + sqrtf(8.0f * (float)p + 1.0f)) * 0.5f);
    while (i * (i - 1) / 2 > p) --i;
    while ((i + 1) * i / 2 <= p) ++i;
    j = p - i * (i - 1) / 2;
}

__launch_bounds__(NTHREADS)
__global__ void getq_kernel(const float* __restrict__ coords,
                            const float* __restrict__ Fin,
                            const float* __restrict__ Ein,
                            float* __restrict__ out)
{
    __shared__ float sChunk[NRP * CS];  // A-chunk [96 rows x 64 k], padded stride
    __shared__ float sG[NRP * GS];      // Gram matrix G = A*A^T (96x96 padded)
    __shared__ float sVec[NRP];         // FE vector -> z -> y (in place)
    __shared__ float sCoord[NRP];       // 90 floats of coords for this sample
    __shared__ float sLam;

    const int tid  = threadIdx.x;
    const int lane = tid & 31;
    const int wave = tid >> 5;          // 0..5
    const int b    = blockIdx.x;

    // ---- stage coords and FE = [F.flatten(), E] ----
    for (int i = tid; i < 90; i += NTHREADS) {
        sCoord[i] = coords[b * 90 + i];
        sVec[i]   = Fin[b * 90 + i];
    }
    if (tid == 0) { sVec[90] = Ein[b]; }
    for (int i = 91 + tid; i < NRP; i += NTHREADS) sVec[i] = 0.0f;
    __syncthreads();

    // ---- G = A * A^T via V_WMMA_F32_16X16X4_F32, chunked over K ----
    v8f acc[6];
    const v8f vzero = {0.f, 0.f, 0.f, 0.f, 0.f, 0.f, 0.f, 0.f};
#pragma unroll
    for (int t = 0; t < 6; ++t) acc[t] = vzero;

    const int l15   = lane & 15;
    const int khalf = (lane >> 4) << 1;       // lanes 16-31 hold k0+2,k0+3
    const int arow  = wave * 16 + l15;        // this wave's A-tile row

    for (int c = 0; c < NCHUNK; ++c) {
        const int kbase = c * KC;

        // zero the chunk (padded rows/columns stay zero)
        for (int i = tid; i < NRP * KC; i += NTHREADS) {
            sChunk[(i >> 6) * CS + (i & 63)] = 0.0f;
        }
        __syncthreads();

        // scatter the 7 nonzeros of each pair column p
        for (int pc = tid; pc < KC; pc += NTHREADS) {
            const int p = kbase + pc;
            if (p < NPAIR) {
                int ai, aj;
                pair_from_index(p, ai, aj);
                const float dx = sCoord[3 * ai + 0] - sCoord[3 * aj + 0];
                const float dy = sCoord[3 * ai + 1] - sCoord[3 * aj + 1];
                const float dz = sCoord[3 * ai + 2] - sCoord[3 * aj + 2];
                const float invr = 1.0f / sqrtf(dx * dx + dy * dy + dz * dz);
                const float ux = dx * invr, uy = dy * invr, uz = dz * invr;
                sChunk[(3 * ai + 0) * CS + pc] =  ux;
                sChunk[(3 * ai + 1) * CS + pc] =  uy;
                sChunk[(3 * ai + 2) * CS + pc] =  uz;
                sChunk[(3 * aj + 0) * CS + pc] = -ux;
                sChunk[(3 * aj + 1) * CS + pc] = -uy;
                sChunk[(3 * aj + 2) * CS + pc] = -uz;
                sChunk[90 * CS + pc]           =  invr;
            }
        }
        __syncthreads();

        // 16 WMMA k-steps over this chunk. For A*A^T the f32 16x16x4 A-frag
        // and B-frag per-lane layouts coincide, so one frag per row-block
        // serves as either operand.
        for (int k0 = 0; k0 < KC; k0 += 4) {
            const int kk = k0 + khalf;
            v2f af;
            af.x = sChunk[arow * CS + kk];
            af.y = sChunk[arow * CS + kk + 1];
            v2f bf[6];
#pragma unroll
            for (int r = 0; r < 6; ++r) {
                bf[r].x = sChunk[(r * 16 + l15) * CS + kk];
                bf[r].y = sChunk[(r * 16 + l15) * CS + kk + 1];
            }
#pragma unroll
            for (int r = 0; r < 6; ++r) {
                acc[r] = __builtin_amdgcn_wmma_f32_16x16x4_f32(
                    false, af, false, bf[r], (short)0, acc[r], false, false);
            }
        }
        __syncthreads();
    }

    // ---- store G tiles: c[v] in lane L -> row = 16*wave + v + 8*(L>=16),
    //      col = 16*J + (L&15) ----
    {
        const int rofs = (lane >> 4) * 8;
#pragma unroll
        for (int J = 0; J < 6; ++J) {
#pragma unroll
            for (int v = 0; v < 8; ++v) {
                const int row = wave * 16 + rofs + v;
                sG[row * GS + J * 16 + (lane & 15)] = acc[J][v];
            }
        }
    }
    __syncthreads();

    // ---- Tikhonov shift: lambda = 1e-6 * trace(G); qs = A^T (G+lI)^-1 FE
    //      converges to pinv since null(A^T)-components are annihilated ----
    if (tid == 0) {
        float tr = 0.0f;
        for (int i = 0; i < NR; ++i) tr += sG[i * GS + i];
        sLam = 1e-6f * tr;
    }
    __syncthreads();
    for (int i = tid; i < NR; i += NTHREADS) sG[i * GS + i] += sLam;
    __syncthreads();

    // ---- cooperative Cholesky, G = L L^T (lower triangle, in place) ----
    for (int k = 0; k < NR; ++k) {
        if (tid == 0) sG[k * GS + k] = sqrtf(fmaxf(sG[k * GS + k], 1e-30f));
        __syncthreads();
        const float dinv = 1.0f / sG[k * GS + k];
        for (int i = k + 1 + tid; i < NR; i += NTHREADS) sG[i * GS + k] *= dinv;
        __syncthreads();
        for (int i = k + 1 + tid; i < NR; i += NTHREADS) {
            const float lik = sG[i * GS + k];
            for (int j = k + 1; j <= i; ++j)
                sG[i * GS + j] -= lik * sG[j * GS + k];
        }
        __syncthreads();
    }

    // ---- forward solve L z = FE (in place in sVec) ----
    for (int k = 0; k < NR; ++k) {
        if (tid == 0) sVec[k] /= sG[k * GS + k];
        __syncthreads();
        const float zk = sVec[k];
        for (int i = k + 1 + tid; i < NR; i += NTHREADS)
            sVec[i] -= sG[i * GS + k] * zk;
        __syncthreads();
    }
    // ---- backward solve L^T y = z (in place) ----
    for (int k = NR - 1; k >= 0; --k) {
        if (tid == 0) sVec[k] /= sG[k * GS + k];
        __syncthreads();
        const float yk = sVec[k];
        for (int i = tid; i < k; i += NTHREADS)
            sVec[i] -= sG[k * GS + i] * yk;
        __syncthreads();
    }

    // ---- qs[p] = u . (y_i - y_j) + invr * y[90]  (A^T y, sparse columns) ----
    for (int p = tid; p < NPAIR; p += NTHREADS) {
        int ai, aj;
        pair_from_index(p, ai, aj);
        const float dx = sCoord[3 * ai + 0] - sCoord[3 * aj + 0];
        const float dy = sCoord[3 * ai + 1] - sCoord[3 * aj + 1];
        const float dz = sCoord[3 * ai + 2] - sCoord[3 * aj + 2];
        const float invr = 1.0f / sqrtf(dx * dx + dy * dy + dz * dz);
        const float q =
            dx * invr * (sVec[3 * ai + 0] - sVec[3 * aj + 0]) +
            dy * invr * (sVec[3 * ai + 1] - sVec[3 * aj + 1]) +
            dz * invr * (sVec[3 * ai + 2] - sVec[3 * aj + 2]) +
            invr * sVec[90];
        out[b * NPAIR + p] = q;
    }
}

extern "C" void kernel_launch(void* const* d_in, const int* in_sizes, int n_in,
                              void* d_out, int out_size, void* d_ws, size_t ws_size,
                              hipStream_t stream) {
    const float* coords = (const float*)d_in[0];
    const float* F      = (const float*)d_in[1];
    const float* E      = (const float*)d_in[2];
    // d_in[3] (atoms_flat) only feeds the discarded _NRF -> unused.
    float* out = (float*)d_out;
    const int batch = in_sizes[2];  // E is [B,1] -> B elements
    (void)n_in; (void)out_size; (void)d_ws; (void)ws_size;
    getq_kernel<<<dim3(batch), dim3(NTHREADS), 0, stream>>>(coords, F, E, out);
}